// GATSequence_59863254172492
// MI455X (gfx1250) — compile-verified
//
#include <hip/hip_runtime.h>
#include <math.h>

#define NEG_SLOPE 0.2f

typedef __attribute__((ext_vector_type(2))) float v2f;
typedef __attribute__((ext_vector_type(8))) float v8f;

// ---------------------------------------------------------------------------
// WMMA fp32 GEMM:  C[M, NCT*16] = act(A[M, K]) @ W[K, NCT*16]
// One wave -> 16 rows x all columns. Uses V_WMMA_F32_16X16X4_F32.
// A-frag layout (16x4 f32): lanes 0-15 hold M=lane, VGPR0=K+0, VGPR1=K+1;
//                           lanes 16-31 hold M=lane-16, VGPR0=K+2, VGPR1=K+3.
// B-frag (4x16): mirrored; C/D (16x16): VGPR r -> row r (lanes 0-15) / r+8.
// ---------------------------------------------------------------------------
template<int K, int NCT, bool RELU_IN>
__global__ __launch_bounds__(128)
void wmma_gemm(const float* __restrict__ A, const float* __restrict__ W,
               float* __restrict__ C, int M) {
  constexpr int NC = NCT * 16;
  const int lane = threadIdx.x & 31;
  const int wave = blockIdx.x * (blockDim.x >> 5) + (threadIdx.x >> 5);
  const int row0 = wave * 16;
  if (row0 >= M) return;                       // wave-uniform guard (EXEC all-1 for WMMA)
  const int nl   = lane & 15;
  const int koff = (lane < 16) ? 0 : 2;
  const int arow = row0 + nl;

  v8f acc[NCT];
#pragma unroll
  for (int t = 0; t < NCT; ++t)
#pragma unroll
    for (int r = 0; r < 8; ++r) acc[t][r] = 0.0f;

  for (int k = 0; k < K; k += 4) {
    float a0 = A[(size_t)arow * K + k + koff];
    float a1 = A[(size_t)arow * K + k + koff + 1];
    if (RELU_IN) { a0 = fmaxf(a0, 0.0f); a1 = fmaxf(a1, 0.0f); }
    v2f av; av.x = a0; av.y = a1;
#pragma unroll
    for (int t = 0; t < NCT; ++t) {
      v2f bv;
      bv.x = W[(k + koff)     * NC + t * 16 + nl];
      bv.y = W[(k + koff + 1) * NC + t * 16 + nl];
      acc[t] = __builtin_amdgcn_wmma_f32_16x16x4_f32(
          /*neg_a=*/false, av, /*neg_b=*/false, bv,
          /*c_mod=*/(short)0, acc[t], /*reuse_a=*/false, /*reuse_b=*/false);
    }
  }

  const int rbase = row0 + ((lane < 16) ? 0 : 8);
#pragma unroll
  for (int t = 0; t < NCT; ++t)
#pragma unroll
    for (int r = 0; r < 8; ++r)
      C[(size_t)(rbase + r) * NC + t * 16 + nl] = acc[t][r];
}

// ---------------------------------------------------------------------------
// alpha_s[n,h] = sum_c h[n,h,c]*a_src[h,c] ; same for alpha_d
// ---------------------------------------------------------------------------
__global__ void alpha_kernel(const float* __restrict__ h,
                             const float* __restrict__ a_src,
                             const float* __restrict__ a_dst,
                             float* __restrict__ as, float* __restrict__ ad,
                             int Nn, int H, int Cc) {
  int i = blockIdx.x * blockDim.x + threadIdx.x;   // i = n*H + head
  if (i >= Nn * H) return;
  int n = i / H, hd = i - n * H;
  const float* hp  = h + ((size_t)n * H + hd) * Cc;
  const float* asw = a_src + hd * Cc;
  const float* adw = a_dst + hd * Cc;
  float s = 0.f, d = 0.f;
  for (int c = 0; c < Cc; ++c) { float v = hp[c]; s += v * asw[c]; d += v * adw[c]; }
  as[i] = s; ad[i] = d;
}

// m=-inf, den=0, out[n, j] = bias[j]
__global__ void init_attn(float* __restrict__ m, float* __restrict__ den,
                          float* __restrict__ out, const float* __restrict__ bias,
                          int NH, int NHC, int HC) {
  int i = blockIdx.x * blockDim.x + threadIdx.x;
  if (i < NH) { m[i] = -INFINITY; den[i] = 0.0f; }
  if (i < NHC) out[i] = bias[i % HC];
}

__device__ __forceinline__ float leaky(float t) {
  return (t > 0.0f) ? t : NEG_SLOPE * t;
}

// float atomic max via sign-bit trick (monotone int ordering)
__device__ __forceinline__ void atomicMaxF32(float* addr, float v) {
  int vi = __float_as_int(v);
  if (vi >= 0) atomicMax((int*)addr, vi);
  else         atomicMin((unsigned int*)addr, (unsigned int)vi);
}

// pass 1: per-target running max of leaky_relu(as[src]+ad[dst])
__global__ void edge_max(const int* __restrict__ src, const int* __restrict__ dst,
                         const float* __restrict__ as, const float* __restrict__ ad,
                         float* __restrict__ m, int E, int H) {
  int i = blockIdx.x * blockDim.x + threadIdx.x;
  if (i >= E * H) return;
  int e = i / H, hd = i - e * H;
  int s = src[e], d = dst[e];
  float t = leaky(as[s * H + hd] + ad[d * H + hd]);
  atomicMaxF32(&m[d * H + hd], t);
}

// pass 2: denom[dst] += exp(t - m[dst])
__global__ void edge_expsum(const int* __restrict__ src, const int* __restrict__ dst,
                            const float* __restrict__ as, const float* __restrict__ ad,
                            const float* __restrict__ m, float* __restrict__ den,
                            int E, int H) {
  int i = blockIdx.x * blockDim.x + threadIdx.x;
  if (i >= E * H) return;
  int e = i / H, hd = i - e * H;
  int s = src[e], d = dst[e];
  float t = leaky(as[s * H + hd] + ad[d * H + hd]);
  atomicAdd(&den[d * H + hd], __expf(t - m[d * H + hd]));
}

// pass 3: out[dst,h,:] += alpha * h[src,h,:]
__global__ void edge_agg(const int* __restrict__ src, const int* __restrict__ dst,
                         const float* __restrict__ as, const float* __restrict__ ad,
                         const float* __restrict__ m, const float* __restrict__ den,
                         const float* __restrict__ h, float* __restrict__ out,
                         int E, int H, int Cc) {
  int i = blockIdx.x * blockDim.x + threadIdx.x;
  if (i >= E * H) return;
  int e = i / H, hd = i - e * H;
  int s = src[e], d = dst[e];
  float t = leaky(as[s * H + hd] + ad[d * H + hd]);
  float alpha = __expf(t - m[d * H + hd]) / (den[d * H + hd] + 1e-16f);
  const float* hs = h   + ((size_t)s * H + hd) * Cc;
  float*       op = out + ((size_t)d * H + hd) * Cc;
  for (int c = 0; c < Cc; ++c) atomicAdd(&op[c], alpha * hs[c]);
}

// final: out[n,:2] = concat_g(o2[g][n]) @ Wf + bf
__global__ void final_linear(const float* __restrict__ o2, const float* __restrict__ Wf,
                             const float* __restrict__ bf, float* __restrict__ out, int Nn) {
  int n = blockIdx.x * blockDim.x + threadIdx.x;
  if (n >= Nn) return;
  float a0 = bf[0], a1 = bf[1];
  for (int g = 0; g < 5; ++g) {
    const float* p = o2 + ((size_t)g * Nn + n) * 16;
#pragma unroll
    for (int c = 0; c < 16; ++c) {
      float v = p[c];
      int k = g * 16 + c;
      a0 += v * Wf[k * 2 + 0];
      a1 += v * Wf[k * 2 + 1];
    }
  }
  out[n * 2 + 0] = a0;
  out[n * 2 + 1] = a1;
}

// ---------------------------------------------------------------------------
extern "C" void kernel_launch(void* const* d_in, const int* in_sizes, int n_in,
                              void* d_out, int out_size, void* d_ws, size_t ws_size,
                              hipStream_t stream) {
  const float* x    = (const float*)d_in[0];   // [G,N,128]
  const int*   ei   = (const int*)  d_in[1];   // [G,2,E]
  const float* W1   = (const float*)d_in[2];   // [128,64]
  const float* as1w = (const float*)d_in[3];   // [8,8]
  const float* ad1w = (const float*)d_in[4];   // [8,8]
  const float* b1   = (const float*)d_in[5];   // [64]
  const float* W2   = (const float*)d_in[6];   // [64,16]
  const float* as2w = (const float*)d_in[7];   // [1,16]
  const float* ad2w = (const float*)d_in[8];   // [1,16]
  const float* b2   = (const float*)d_in[9];   // [16]
  const float* Wf   = (const float*)d_in[10];  // [80,2]
  const float* bf   = (const float*)d_in[11];  // [2]

  const int G = 5, F = 128;
  const int Nn = in_sizes[0] / (G * F);        // 50000
  const int E  = in_sizes[1] / (G * 2);        // 850000

  // workspace layout (floats)
  float* ws    = (float*)d_ws;
  float* o2all = ws;                                   // G*N*16
  float* h1    = o2all + (size_t)G * Nn * 16;          // N*64 (layer-2 h aliases this)
  float* o1    = h1 + (size_t)Nn * 64;                 // N*64
  float* asb   = o1 + (size_t)Nn * 64;                 // N*8  (layer-2 aliases)
  float* adb   = asb + (size_t)Nn * 8;
  float* mb    = adb + (size_t)Nn * 8;
  float* db    = mb + (size_t)Nn * 8;
  float* h2    = h1;                                   // alias: h1 dead after layer-1 agg
  float* as2b  = asb, *ad2b = adb, *m2b = mb, *d2b = db;

  const int TB = 256;
  const int waves = Nn / 16;                           // 3125, exact
  const int gemmBlocks = (waves + 3) / 4;              // 4 waves / 128-thread block

  for (int g = 0; g < G; ++g) {
    const float* xg   = x + (size_t)g * Nn * F;
    const int*   srcg = ei + (size_t)g * 2 * E;
    const int*   dstg = srcg + E;
    float*       o2g  = o2all + (size_t)g * Nn * 16;

    // ---- layer 1 (H=8, C=8) ----
    wmma_gemm<128, 4, false><<<gemmBlocks, 128, 0, stream>>>(xg, W1, h1, Nn);
    alpha_kernel<<<(Nn * 8 + TB - 1) / TB, TB, 0, stream>>>(h1, as1w, ad1w, asb, adb, Nn, 8, 8);
    init_attn<<<(Nn * 64 + TB - 1) / TB, TB, 0, stream>>>(mb, db, o1, b1, Nn * 8, Nn * 64, 64);
    edge_max   <<<(E * 8 + TB - 1) / TB, TB, 0, stream>>>(srcg, dstg, asb, adb, mb, E, 8);
    edge_expsum<<<(E * 8 + TB - 1) / TB, TB, 0, stream>>>(srcg, dstg, asb, adb, mb, db, E, 8);
    edge_agg   <<<(E * 8 + TB - 1) / TB, TB, 0, stream>>>(srcg, dstg, asb, adb, mb, db, h1, o1, E, 8, 8);

    // ---- layer 2 (H=1, C=16), relu applied to o1 inside GEMM ----
    wmma_gemm<64, 1, true><<<gemmBlocks, 128, 0, stream>>>(o1, W2, h2, Nn);
    alpha_kernel<<<(Nn + TB - 1) / TB, TB, 0, stream>>>(h2, as2w, ad2w, as2b, ad2b, Nn, 1, 16);
    init_attn<<<(Nn * 16 + TB - 1) / TB, TB, 0, stream>>>(m2b, d2b, o2g, b2, Nn, Nn * 16, 16);
    edge_max   <<<(E + TB - 1) / TB, TB, 0, stream>>>(srcg, dstg, as2b, ad2b, m2b, E, 1);
    edge_expsum<<<(E + TB - 1) / TB, TB, 0, stream>>>(srcg, dstg, as2b, ad2b, m2b, d2b, E, 1);
    edge_agg   <<<(E + TB - 1) / TB, TB, 0, stream>>>(srcg, dstg, as2b, ad2b, m2b, d2b, h2, o2g, E, 1, 16);
  }

  final_linear<<<(Nn + TB - 1) / TB, TB, 0, stream>>>(o2all, Wf, bf, (float*)d_out, Nn);
}